// Model_68410239091111
// MI455X (gfx1250) — compile-verified
//
#include <hip/hip_runtime.h>

#define N_NODES 50000
#define N_EDGES 1600000
#define F_IN    8
#define F_OUT   12
#define PERIODS 6
#define C2      24   // packed conv outputs per node: [z(12) | h(12)]

typedef __attribute__((ext_vector_type(2))) float v2f;
typedef __attribute__((ext_vector_type(8))) float v8f;

// ---------------------------------------------------------------------------
// 1) init: acc = 0, deg = 1.0 (self-loop weight)
// ---------------------------------------------------------------------------
__global__ void k_init(float* __restrict__ deg, float* __restrict__ acc, int accTotal) {
  int i = blockIdx.x * blockDim.x + threadIdx.x;
  if (i < accTotal) acc[i] = 0.0f;
  if (i < N_NODES)  deg[i] = 1.0f;
}

// ---------------------------------------------------------------------------
// 2) degree accumulation over dst
// ---------------------------------------------------------------------------
__global__ void k_degree(const int* __restrict__ dst, const float* __restrict__ ew,
                         float* __restrict__ deg) {
  int e = blockIdx.x * blockDim.x + threadIdx.x;
  if (e >= N_EDGES) return;
  unsafeAtomicAdd(&deg[dst[e]], ew[e]);
}

// ---------------------------------------------------------------------------
// 3) dinv = deg > 0 ? 1/sqrt(deg) : 0
// ---------------------------------------------------------------------------
__global__ void k_dinv(const float* __restrict__ deg, float* __restrict__ dinv) {
  int i = blockIdx.x * blockDim.x + threadIdx.x;
  if (i >= N_NODES) return;
  float d = deg[i];
  dinv[i] = d > 0.0f ? rsqrtf(d) : 0.0f;
}

// ---------------------------------------------------------------------------
// 4) dense feature transform via WMMA (f32, exact):
//    Hc[p][n][0:24] = x[n,:,p] @ [Wz | Wh]   (K = 8, chained 2x K=4 WMMA)
//    One wave handles a 16-node tile for one period.
// ---------------------------------------------------------------------------
__global__ void k_gemm_wmma(const float* __restrict__ x,   // [N, F_IN, PERIODS]
                            const float* __restrict__ Wz,  // [F_IN, F_OUT]
                            const float* __restrict__ Wh,  // [F_IN, F_OUT]
                            float* __restrict__ Hc)        // [PERIODS, N, C2]
{
  const int tilesN = (N_NODES + 15) / 16;
  int wavesPerBlock = blockDim.x >> 5;
  int wave = blockIdx.x * wavesPerBlock + (threadIdx.x >> 5);
  if (wave >= tilesN * PERIODS) return;
  int tile = wave / PERIODS;
  int p    = wave - tile * PERIODS;

  int lane = threadIdx.x & 31;
  int m    = lane & 15;     // M row (A) / N col (B,D)
  int kh   = lane >> 4;     // selects K pair within a K=4 chunk

  int  node   = tile * 16 + m;
  bool nvalid = node < N_NODES;
  int  nodec  = nvalid ? node : (N_NODES - 1);
  const float* xrow = x + (size_t)nodec * (F_IN * PERIODS) + p;

  // A 16x4 f32 layout: VGPR0 = K {0|2}, VGPR1 = K {1|3} by lane half.
  int k0 = kh * 2;
  v2f a0, a1;
  {
    float a00 = xrow[(k0 + 0) * PERIODS];
    float a01 = xrow[(k0 + 1) * PERIODS];
    float a10 = xrow[(k0 + 4) * PERIODS];
    float a11 = xrow[(k0 + 5) * PERIODS];
    a0[0] = nvalid ? a00 : 0.0f;  a0[1] = nvalid ? a01 : 0.0f;
    a1[0] = nvalid ? a10 : 0.0f;  a1[1] = nvalid ? a11 : 0.0f;
  }

  // Two 16-wide N tiles covering the 24 packed columns [Wz | Wh].
  for (int t = 0; t < 2; ++t) {
    int  col    = t * 16 + m;
    bool cvalid = col < C2;
    int  colc   = cvalid ? col : 0;
    const float* Wp = (colc < F_OUT) ? (Wz + colc) : (Wh + (colc - F_OUT));

    v2f b0, b1;
    {
      float b00 = Wp[(k0 + 0) * F_OUT];
      float b01 = Wp[(k0 + 1) * F_OUT];
      float b10 = Wp[(k0 + 4) * F_OUT];
      float b11 = Wp[(k0 + 5) * F_OUT];
      b0[0] = cvalid ? b00 : 0.0f;  b0[1] = cvalid ? b01 : 0.0f;
      b1[0] = cvalid ? b10 : 0.0f;  b1[1] = cvalid ? b11 : 0.0f;
    }

    v8f c = {};
    c = __builtin_amdgcn_wmma_f32_16x16x4_f32(false, a0, false, b0, (short)0, c, false, false);
    c = __builtin_amdgcn_wmma_f32_16x16x4_f32(false, a1, false, b1, (short)0, c, false, false);

    if (cvalid) {
      float* outp = Hc + (size_t)p * N_NODES * C2 + col;
      #pragma unroll
      for (int r = 0; r < 8; ++r) {
        int nr = tile * 16 + r + kh * 8;   // D: M = r + 8*(lane>=16)
        if (nr < N_NODES) outp[(size_t)nr * C2] = c[r];
      }
    }
  }
}

// ---------------------------------------------------------------------------
// 5) edge scatter: acc[p][dst] += norm * Hc[p][src]   (fire-and-forget f32
//    atomics; all hot data resident in 192MB L2)
// ---------------------------------------------------------------------------
__global__ void k_edge(const int* __restrict__ src, const int* __restrict__ dst,
                       const float* __restrict__ ew, const float* __restrict__ dinv,
                       const float* __restrict__ Hc, float* __restrict__ acc)
{
  int e = blockIdx.x * blockDim.x + threadIdx.x;
  if (e >= N_EDGES) return;
  int p = blockIdx.y;

  int s = src[e], d = dst[e];
  float norm = dinv[s] * ew[e] * dinv[d];

  const float4* hrow = (const float4*)(Hc + ((size_t)p * N_NODES + s) * C2);
  float* arow = acc + ((size_t)p * N_NODES + d) * C2;

  #pragma unroll
  for (int v = 0; v < C2 / 4; ++v) {
    float4 h = hrow[v];
    unsafeAtomicAdd(&arow[v * 4 + 0], norm * h.x);
    unsafeAtomicAdd(&arow[v * 4 + 1], norm * h.y);
    unsafeAtomicAdd(&arow[v * 4 + 2], norm * h.z);
    unsafeAtomicAdd(&arow[v * 4 + 3], norm * h.w);
  }
}

// ---------------------------------------------------------------------------
// 6) finalize: add self-loop + bias, gates, attention mix, relu, projection
// ---------------------------------------------------------------------------
__global__ void k_finalize(const float* __restrict__ Hc, const float* __restrict__ acc,
                           const float* __restrict__ dinv, const float* __restrict__ att,
                           const float* __restrict__ bz,  const float* __restrict__ bh,
                           const float* __restrict__ lzW, const float* __restrict__ lzb,
                           const float* __restrict__ lhW, const float* __restrict__ lhb,
                           const float* __restrict__ linW, const float* __restrict__ linb,
                           float* __restrict__ out)
{
  __shared__ float sAz[F_OUT * F_OUT], sAh[F_OUT * F_OUT];
  __shared__ float sbz[F_OUT], sbh[F_OUT], szb[F_OUT], shb[F_OUT];
  __shared__ float sLin[PERIODS * F_OUT], sLb[PERIODS], sProb[PERIODS];

  for (int i = threadIdx.x; i < F_OUT * F_OUT; i += blockDim.x) {
    int j = i / F_OUT, k = i % F_OUT;
    sAz[i] = lzW[j * (2 * F_OUT) + k];   // only first 12 cols matter (H0 == 0)
    sAh[i] = lhW[j * (2 * F_OUT) + k];
  }
  for (int i = threadIdx.x; i < F_OUT; i += blockDim.x) {
    sbz[i] = bz[i]; sbh[i] = bh[i]; szb[i] = lzb[i]; shb[i] = lhb[i];
  }
  for (int i = threadIdx.x; i < PERIODS * F_OUT; i += blockDim.x) sLin[i] = linW[i];
  if (threadIdx.x < PERIODS) sLb[threadIdx.x] = linb[threadIdx.x];
  if (threadIdx.x == 0) {
    float mx = att[0];
    for (int i = 1; i < PERIODS; ++i) mx = fmaxf(mx, att[i]);
    float s = 0.0f, e[PERIODS];
    for (int i = 0; i < PERIODS; ++i) { e[i] = __expf(att[i] - mx); s += e[i]; }
    for (int i = 0; i < PERIODS; ++i) sProb[i] = e[i] / s;
  }
  __syncthreads();

  int n = blockIdx.x * blockDim.x + threadIdx.x;
  if (n >= N_NODES) return;

  float sn = dinv[n]; sn = sn * sn;   // self-loop norm

  float H[F_OUT];
  #pragma unroll
  for (int j = 0; j < F_OUT; ++j) H[j] = 0.0f;

  for (int p = 0; p < PERIODS; ++p) {
    const float* hrow = Hc  + ((size_t)p * N_NODES + n) * C2;
    const float* arow = acc + ((size_t)p * N_NODES + n) * C2;
    float cz[F_OUT], ch[F_OUT];
    #pragma unroll
    for (int i = 0; i < F_OUT; ++i) {
      cz[i] = arow[i]         + sn * hrow[i]         + sbz[i];
      ch[i] = arow[F_OUT + i] + sn * hrow[F_OUT + i] + sbh[i];
    }
    float pw = sProb[p];
    #pragma unroll
    for (int j = 0; j < F_OUT; ++j) {
      float az = szb[j], ah = shb[j];
      #pragma unroll
      for (int i = 0; i < F_OUT; ++i) {
        az = fmaf(cz[i], sAz[j * F_OUT + i], az);
        ah = fmaf(ch[i], sAh[j * F_OUT + i], ah);
      }
      float Z = 1.0f / (1.0f + __expf(-az));
      float T = tanhf(ah);
      H[j] += pw * (1.0f - Z) * T;   // Z*H0 + (1-Z)*Ht with H0 = 0
    }
  }

  float* o = out + (size_t)n * PERIODS;
  #pragma unroll
  for (int t = 0; t < PERIODS; ++t) {
    float v = sLb[t];
    #pragma unroll
    for (int j = 0; j < F_OUT; ++j)
      v = fmaf(fmaxf(H[j], 0.0f), sLin[t * F_OUT + j], v);
    o[t] = v;
  }
}

// ---------------------------------------------------------------------------
static inline size_t align256(size_t x) { return (x + 255) & ~(size_t)255; }

extern "C" void kernel_launch(void* const* d_in, const int* in_sizes, int n_in,
                              void* d_out, int out_size, void* d_ws, size_t ws_size,
                              hipStream_t stream) {
  const float* x    = (const float*)d_in[0];
  const int*   ei   = (const int*)  d_in[1];   // [2, E]
  const float* ew   = (const float*)d_in[2];
  const float* att  = (const float*)d_in[3];
  const float* Wz   = (const float*)d_in[4];
  const float* bz   = (const float*)d_in[5];
  // d_in[6], d_in[7] (Wr, br) dead: H0==0 makes R unused
  const float* Wh   = (const float*)d_in[8];
  const float* bh   = (const float*)d_in[9];
  const float* lzW  = (const float*)d_in[10];
  const float* lzb  = (const float*)d_in[11];
  // d_in[12], d_in[13] (lrW, lrb) dead
  const float* lhW  = (const float*)d_in[14];
  const float* lhb  = (const float*)d_in[15];
  const float* linW = (const float*)d_in[16];
  const float* linb = (const float*)d_in[17];
  float* out = (float*)d_out;

  const int* esrc = ei;
  const int* edst = ei + N_EDGES;

  // workspace carve-out
  char* ws = (char*)d_ws;
  size_t o = 0;
  float* deg  = (float*)(ws + o); o += align256((size_t)N_NODES * 4);
  float* dinv = (float*)(ws + o); o += align256((size_t)N_NODES * 4);
  float* Hc   = (float*)(ws + o); o += align256((size_t)PERIODS * N_NODES * C2 * 4);
  float* acc  = (float*)(ws + o); o += align256((size_t)PERIODS * N_NODES * C2 * 4);
  (void)ws_size; (void)n_in; (void)in_sizes; (void)out_size;

  const int accTotal = PERIODS * N_NODES * C2;

  k_init<<<(accTotal + 255) / 256, 256, 0, stream>>>(deg, acc, accTotal);
  k_degree<<<(N_EDGES + 255) / 256, 256, 0, stream>>>(edst, ew, deg);
  k_dinv<<<(N_NODES + 255) / 256, 256, 0, stream>>>(deg, dinv);

  {
    const int tilesN = (N_NODES + 15) / 16;
    int waves = tilesN * PERIODS;
    int wavesPerBlock = 4;                 // 128 threads = 4 wave32
    int blocks = (waves + wavesPerBlock - 1) / wavesPerBlock;
    k_gemm_wmma<<<blocks, 128, 0, stream>>>(x, Wz, Wh, Hc);
  }

  {
    dim3 grid((N_EDGES + 255) / 256, PERIODS);
    k_edge<<<grid, 256, 0, stream>>>(esrc, edst, ew, dinv, Hc, acc);
  }

  k_finalize<<<(N_NODES + 255) / 256, 256, 0, stream>>>(
      Hc, acc, dinv, att, bz, bh, lzW, lzb, lhW, lhb, linW, linb, out);
}